// Generalized_loss_eff_fg_68049461838568
// MI455X (gfx1250) — compile-verified
//
#include <hip/hip_runtime.h>

typedef float v2f __attribute__((ext_vector_type(2)));
typedef float v8f __attribute__((ext_vector_type(8)));

#define NS   2048          // number of sample rows (K dimension of the Gram matmuls)
#define L    128           // series length (M and N of the Gram matrices)
#define FGV  2.0f
#define NTILE  8           // L / 16
#define NPAIRS 36          // NTILE*(NTILE+1)/2 upper-triangular 16x16 tile pairs

// ws layout (floats): ws[0] = loss accumulator, ws[1] = sum(f) accumulator

__global__ void k_init(float* ws) {
  ws[0] = 0.0f;
  ws[1] = 0.0f;
}

// ---------------------------------------------------------------------------
// Pass 1: the non-bilinear p=1 term  sum w1[i,k]*|d_i - d_{i+k}|*f_i*f_{i+k}
// (k=0 uses |d_i|*f_i), plus the global sum of f. One block per sample row n;
// the row of d and f lives in LDS so the k-loop is an LDS broadcast.
// ---------------------------------------------------------------------------
__global__ __launch_bounds__(L) void k_p1(const float* __restrict__ yt,
                                          const float* __restrict__ yp,
                                          const float* __restrict__ yd,
                                          const float* __restrict__ w,   // weights: w1 = w, w2 = w + L*L
                                          float* __restrict__ ws) {
  __shared__ float sd[L];
  __shared__ float sf[L];
  __shared__ float red[L];

  const int n = blockIdx.x;
  const int i = threadIdx.x;
  const int base = n * L;

  const float d = yt[base + i] - yp[base + i];
  const float f = (fabsf(yd[base + i]) > FGV) ? 0.0f : 1.0f;
  sd[i] = d;
  sf[i] = f;
  __syncthreads();

  const float* w1row = w + i * L;
  __builtin_prefetch(w1row, 0, 3);          // global_prefetch_b8 on the weight row

  float acc = w1row[0] * fabsf(d) * f;      // k == 0 term
  const int kmax = L - i;                   // valid: i + k < L
  for (int k = 1; k < kmax; ++k) {
    const int j = i + k;
    acc += w1row[k] * fabsf(d - sd[j]) * (f * sf[j]);
  }

  // block reduction of the loss partial
  red[i] = acc;
  __syncthreads();
  for (int s = L / 2; s > 0; s >>= 1) {
    if (i < s) red[i] += red[i + s];
    __syncthreads();
  }
  if (i == 0) atomicAdd(&ws[0], red[0]);
  __syncthreads();

  // block reduction of sum(f)
  red[i] = f;
  __syncthreads();
  for (int s = L / 2; s > 0; s >>= 1) {
    if (i < s) red[i] += red[i + s];
    __syncthreads();
  }
  if (i == 0) atomicAdd(&ws[1], red[0]);
}

// ---------------------------------------------------------------------------
// Pass 2: the bilinear p=2 term via fp32 WMMA Gram matrices.
//   (d_i-d_j)^2 f_i f_j = v_i w_j - 2 u_i u_j + w_i v_j,
//   u = d*f, v = d^2*f, w = f, summed over n  ->  AtB Grams with K = NS.
// One wave (32 lanes, blockDim=32) per upper-triangular tile pair (ti<=tj):
// accumulates G_vw, G_wv, G_uu tiles with V_WMMA_F32_16X16X4_F32, then
// contracts in-register against w2[i, j-i] (j>i) plus the diagonal
// k=0 term w2[i,0]*G_uu[i,i], and atomically adds the partial.
// ---------------------------------------------------------------------------
__global__ __launch_bounds__(32) void k_p2(const float* __restrict__ yt,
                                           const float* __restrict__ yp,
                                           const float* __restrict__ yd,
                                           const float* __restrict__ w,
                                           float* __restrict__ ws) {
  const float* __restrict__ w2 = w + L * L;

  // map blockIdx.x -> (ti, tj) with ti <= tj over an 8x8 tile grid
  int p = blockIdx.x;
  int ti = 0;
  while (p >= NTILE - ti) { p -= NTILE - ti; ++ti; }
  const int tj = ti + p;

  const int lane = threadIdx.x & 31;
  const int m    = lane & 15;     // M index for A rows / N index for B cols / col of C
  const int half = lane >> 4;     // lane half selects K pair {0,1} vs {2,3}

  const int ic = ti * 16 + m;     // A-operand column in the data (i side)
  const int jc = tj * 16 + m;     // B-operand column in the data (j side)

  v8f g_vw = {};                  // v^T w   -> G1[i,j]
  v8f g_wv = {};                  // w^T v   -> G1[j,i] seen as [i,j]
  v8f g_uu = {};                  // u^T u   -> G2[i,j]

  for (int k0 = 0; k0 < NS; k0 += 4) {
    const int r0 = k0 + 2 * half;     // this lane-half supplies K = r0, r0+1
    const int r1 = r0 + 1;

    // A side (column ic): u, v, w on the fly
    const float da0 = yt[r0 * L + ic] - yp[r0 * L + ic];
    const float fa0 = (fabsf(yd[r0 * L + ic]) > FGV) ? 0.0f : 1.0f;
    const float da1 = yt[r1 * L + ic] - yp[r1 * L + ic];
    const float fa1 = (fabsf(yd[r1 * L + ic]) > FGV) ? 0.0f : 1.0f;
    const v2f aU = { da0 * fa0,        da1 * fa1 };
    const v2f aV = { da0 * da0 * fa0,  da1 * da1 * fa1 };
    const v2f aW = { fa0,              fa1 };

    // B side (column jc)
    const float db0 = yt[r0 * L + jc] - yp[r0 * L + jc];
    const float fb0 = (fabsf(yd[r0 * L + jc]) > FGV) ? 0.0f : 1.0f;
    const float db1 = yt[r1 * L + jc] - yp[r1 * L + jc];
    const float fb1 = (fabsf(yd[r1 * L + jc]) > FGV) ? 0.0f : 1.0f;
    const v2f bU = { db0 * fb0,        db1 * fb1 };
    const v2f bV = { db0 * db0 * fb0,  db1 * db1 * fb1 };
    const v2f bW = { fb0,              fb1 };

    g_vw = __builtin_amdgcn_wmma_f32_16x16x4_f32(false, aV, false, bW, (short)0, g_vw, false, false);
    g_wv = __builtin_amdgcn_wmma_f32_16x16x4_f32(false, aW, false, bV, (short)0, g_wv, false, false);
    g_uu = __builtin_amdgcn_wmma_f32_16x16x4_f32(false, aU, false, bU, (short)0, g_uu, false, false);
  }

  // In-register contraction against w2. C/D layout: element r of the v8f in
  // this lane is (row = r + 8*half, col = m) of the 16x16 tile.
  float partial = 0.0f;
  for (int r = 0; r < 8; ++r) {
    const int ig = ti * 16 + r + 8 * half;
    const int jg = tj * 16 + m;
    if (jg > ig) {
      partial += w2[ig * L + (jg - ig)] * (g_vw[r] + g_wv[r] - 2.0f * g_uu[r]);
    } else if (jg == ig) {
      partial += w2[ig * L] * g_uu[r];   // k==0 p=2 term: w2[i,0]*sum_n d_i^2 f_i
    }
  }
  atomicAdd(&ws[0], partial);
}

__global__ void k_fin(const float* ws, float* out) {
  // loss / L / (N*mean(f)) == loss / sum(f)
  out[0] = ws[0] / ws[1];
}

extern "C" void kernel_launch(void* const* d_in, const int* in_sizes, int n_in,
                              void* d_out, int out_size, void* d_ws, size_t ws_size,
                              hipStream_t stream) {
  const float* yt = (const float*)d_in[0];   // y_true   (N, L) fp32
  const float* yp = (const float*)d_in[1];   // y_pred   (N, L) fp32
  const float* yd = (const float*)d_in[2];   // y_diff   (N, L) fp32
  const float* w  = (const float*)d_in[3];   // weights  (2, L, L) fp32
  float* ws  = (float*)d_ws;
  float* out = (float*)d_out;

  k_init<<<1, 1, 0, stream>>>(ws);
  k_p1 <<<NS, L, 0, stream>>>(yt, yp, yd, w, ws);
  k_p2 <<<NPAIRS, 32, 0, stream>>>(yt, yp, yd, w, ws);
  k_fin<<<1, 1, 0, stream>>>(ws, out);
}